// DotProductAttention_7997229105272
// MI455X (gfx1250) — compile-verified
//
#include <hip/hip_runtime.h>
#include <hip/hip_bf16.h>

typedef float v2f __attribute__((ext_vector_type(2)));
typedef float v4f __attribute__((ext_vector_type(4)));
typedef float v8f __attribute__((ext_vector_type(8)));

#define B_ 4
#define LQ 4096
#define LK 4096
#define DD 512

__device__ __forceinline__ void st4(float* p, v4f v) {
    v2f lo = {v.x, v.y};
    v2f hi = {v.z, v.w};
    *reinterpret_cast<v2f*>(p)     = lo;
    *reinterpret_cast<v2f*>(p + 2) = hi;
}

// ---------------------------------------------------------------------------
// Kernel 1: S[b,q,k] = sum_d Q[b,q,d] * V[b,k,d]   (NT GEMM, raw scores)
// 128x128 tile / workgroup, KC=32. 16 waves, each wave a 2x2 block of
// 16x16 WMMA subtiles -> 4 WMMAs per 4 operand loads, 32 WMMAs per barrier.
// ---------------------------------------------------------------------------
__global__ __launch_bounds__(512) void attn_qk_kernel(const float* __restrict__ Q,
                                                      const float* __restrict__ V,
                                                      float* __restrict__ S) {
    __shared__ float sQ[128][36];   // stride 36 -> conflict-free b64 WMMA reads
    __shared__ float sV[128][36];

    const int b  = blockIdx.z;
    const int q0 = blockIdx.y * 128;
    const int k0 = blockIdx.x * 128;

    const float* Qb = Q + (size_t)b * LQ * DD + (size_t)q0 * DD;
    const float* Vb = V + (size_t)b * LK * DD + (size_t)k0 * DD;

    const int t    = threadIdx.x;
    const int lane = t & 31;
    const int w    = t >> 5;            // wave 0..15
    const int wm   = w >> 2;            // 0..3 (32-row block)
    const int wn   = w & 3;             // 0..3 (32-col block)
    const int m    = lane & 15;
    const int hi2  = (lane >> 4) << 1;  // 0 | 2
    const int hi8  = (lane >> 4) << 3;  // 0 | 8

    // staging: each array 128 rows x 32 cols; thread does rows rS, rS+64 (v4f)
    const int rS = t >> 3;              // 0..63
    const int cS = (t & 7) << 2;        // 0,4,...,28

    v8f acc00 = {}, acc01 = {}, acc10 = {}, acc11 = {};

    for (int dc = 0; dc < DD; dc += 32) {
        __syncthreads();
        v4f qa = *reinterpret_cast<const v4f*>(Qb + (size_t)rS * DD + dc + cS);
        v4f qb = *reinterpret_cast<const v4f*>(Qb + (size_t)(rS + 64) * DD + dc + cS);
        v4f va = *reinterpret_cast<const v4f*>(Vb + (size_t)rS * DD + dc + cS);
        v4f vb = *reinterpret_cast<const v4f*>(Vb + (size_t)(rS + 64) * DD + dc + cS);
        st4(&sQ[rS][cS], qa);
        st4(&sQ[rS + 64][cS], qb);
        st4(&sV[rS][cS], va);
        st4(&sV[rS + 64][cS], vb);
        __syncthreads();

#pragma unroll
        for (int dd = 0; dd < 8; ++dd) {
            v2f a0 = *reinterpret_cast<const v2f*>(&sQ[wm * 32 + m][dd * 4 + hi2]);
            v2f a1 = *reinterpret_cast<const v2f*>(&sQ[wm * 32 + 16 + m][dd * 4 + hi2]);
            v2f b0 = *reinterpret_cast<const v2f*>(&sV[wn * 32 + m][dd * 4 + hi2]);
            v2f b1 = *reinterpret_cast<const v2f*>(&sV[wn * 32 + 16 + m][dd * 4 + hi2]);
            acc00 = __builtin_amdgcn_wmma_f32_16x16x4_f32(false, a0, false, b0, (short)0, acc00, false, false);
            acc01 = __builtin_amdgcn_wmma_f32_16x16x4_f32(false, a0, false, b1, (short)0, acc01, false, false);
            acc10 = __builtin_amdgcn_wmma_f32_16x16x4_f32(false, a1, false, b0, (short)0, acc10, false, false);
            acc11 = __builtin_amdgcn_wmma_f32_16x16x4_f32(false, a1, false, b1, (short)0, acc11, false, false);
        }
    }

    float* Sp = S + (size_t)b * LQ * LK + (size_t)(q0 + wm * 32) * LK + (k0 + wn * 32);
#pragma unroll
    for (int j = 0; j < 8; ++j) {
        const size_t r0 = (size_t)(j + hi8) * LK;
        const size_t r1 = (size_t)(16 + j + hi8) * LK;
        Sp[r0 + m]      = acc00[j];
        Sp[r0 + 16 + m] = acc01[j];
        Sp[r1 + m]      = acc10[j];
        Sp[r1 + 16 + m] = acc11[j];
    }
}

// ---------------------------------------------------------------------------
// Kernel 2: in-place row softmax over LK (one 256-thread block per row)
// ---------------------------------------------------------------------------
__global__ __launch_bounds__(256) void attn_softmax_kernel(float* __restrict__ S) {
    __shared__ float red[8];
    float* p = S + (size_t)blockIdx.x * LK;

    const int t    = threadIdx.x;
    const int lane = t & 31;
    const int w    = t >> 5;

    float x[16];
#pragma unroll
    for (int i = 0; i < 16; ++i) x[i] = p[t + i * 256];

    float mx = x[0];
#pragma unroll
    for (int i = 1; i < 16; ++i) mx = fmaxf(mx, x[i]);
#pragma unroll
    for (int o = 16; o > 0; o >>= 1) mx = fmaxf(mx, __shfl_xor(mx, o, 32));
    if (lane == 0) red[w] = mx;
    __syncthreads();
    float bm = red[0];
#pragma unroll
    for (int k = 1; k < 8; ++k) bm = fmaxf(bm, red[k]);
    __syncthreads();

    float s = 0.f;
#pragma unroll
    for (int i = 0; i < 16; ++i) {
        x[i] = __expf(x[i] - bm);
        s += x[i];
    }
#pragma unroll
    for (int o = 16; o > 0; o >>= 1) s += __shfl_xor(s, o, 32);
    if (lane == 0) red[w] = s;
    __syncthreads();
    float tot = red[0];
#pragma unroll
    for (int k = 1; k < 8; ++k) tot += red[k];
    const float inv = 1.f / tot;

#pragma unroll
    for (int i = 0; i < 16; ++i) p[t + i * 256] = x[i] * inv;
}

// ---------------------------------------------------------------------------
// Kernel 3: O[b,q,d] = sum_k P[b,q,k] * V[b,k,d]   (NN GEMM)
// 128x64 tile / workgroup, KC=32. Each wave a 2x1 subtile block.
// ---------------------------------------------------------------------------
__global__ __launch_bounds__(512) void attn_pv_kernel(const float* __restrict__ P,
                                                      const float* __restrict__ V,
                                                      float* __restrict__ O) {
    __shared__ float sA[128][36];  // P chunk: 128 q-rows x 32 k
    __shared__ float sB[32][80];   // V chunk:  32 k-rows x 64 d

    const int b  = blockIdx.z;
    const int q0 = blockIdx.y * 128;
    const int d0 = blockIdx.x * 64;

    const float* Pb = P + (size_t)b * LQ * LK + (size_t)q0 * LK;
    const float* Vb = V + (size_t)b * LK * DD;

    const int t    = threadIdx.x;
    const int lane = t & 31;
    const int w    = t >> 5;
    const int wm   = w >> 2;            // 0..3 (32-row block)
    const int wn   = w & 3;             // 0..3 (16-col block)
    const int m    = lane & 15;
    const int hi2  = (lane >> 4) << 1;
    const int hi8  = (lane >> 4) << 3;

    const int rA = t >> 3;              // 0..63
    const int cA = (t & 7) << 2;        // 0..28
    const int rB = t >> 4;              // 0..31
    const int cB = (t & 15) << 2;       // 0..60

    v8f acc0 = {}, acc1 = {};

    for (int kc = 0; kc < LK; kc += 32) {
        __syncthreads();
        v4f pa = *reinterpret_cast<const v4f*>(Pb + (size_t)rA * LK + kc + cA);
        v4f pb = *reinterpret_cast<const v4f*>(Pb + (size_t)(rA + 64) * LK + kc + cA);
        v4f vv = *reinterpret_cast<const v4f*>(Vb + (size_t)(kc + rB) * DD + d0 + cB);
        st4(&sA[rA][cA], pa);
        st4(&sA[rA + 64][cA], pb);
        *reinterpret_cast<v4f*>(&sB[rB][cB]) = vv;   // 320B row stride: 16B aligned
        __syncthreads();

#pragma unroll
        for (int dd = 0; dd < 8; ++dd) {
            v2f a0 = *reinterpret_cast<const v2f*>(&sA[wm * 32 + m][dd * 4 + hi2]);
            v2f a1 = *reinterpret_cast<const v2f*>(&sA[wm * 32 + 16 + m][dd * 4 + hi2]);
            const int kr = dd * 4 + hi2;
            v2f bb;
            bb.x = sB[kr][wn * 16 + m];
            bb.y = sB[kr + 1][wn * 16 + m];
            acc0 = __builtin_amdgcn_wmma_f32_16x16x4_f32(false, a0, false, bb, (short)0, acc0, false, false);
            acc1 = __builtin_amdgcn_wmma_f32_16x16x4_f32(false, a1, false, bb, (short)0, acc1, false, false);
        }
    }

    float* Op = O + (size_t)b * LQ * DD + (size_t)(q0 + wm * 32) * DD + (d0 + wn * 16);
#pragma unroll
    for (int j = 0; j < 8; ++j) {
        Op[(size_t)(j + hi8) * DD + m]      = acc0[j];
        Op[(size_t)(16 + j + hi8) * DD + m] = acc1[j];
    }
}

// ---------------------------------------------------------------------------
extern "C" void kernel_launch(void* const* d_in, const int* in_sizes, int n_in,
                              void* d_out, int out_size, void* d_ws, size_t ws_size,
                              hipStream_t stream) {
    (void)in_sizes; (void)n_in; (void)out_size; (void)d_ws; (void)ws_size;
    const float* Q = (const float*)d_in[0];
    const float* V = (const float*)d_in[1];
    float* ctx  = (float*)d_out;                          // [B, Lq, D]
    float* attn = (float*)d_out + (size_t)B_ * LQ * DD;   // [B, Lq, Lk]

    // 1) raw scores into attn region
    attn_qk_kernel<<<dim3(LK / 128, LQ / 128, B_), 512, 0, stream>>>(Q, V, attn);
    // 2) in-place row softmax
    attn_softmax_kernel<<<B_ * LQ, 256, 0, stream>>>(attn);
    // 3) context = attn @ V
    attn_pv_kernel<<<dim3(DD / 64, LQ / 128, B_), 512, 0, stream>>>(attn, V, ctx);
}